// SelectiveBlock_52974126629486
// MI455X (gfx1250) — compile-verified
//
#include <hip/hip_runtime.h>

// ---------------------------------------------------------------------------
// SelectiveBlock (Mamba-style) for gfx1250 / MI455X, fp32 via V_WMMA_F32_16X16X4_F32
// Shapes: B=4, L=512, EMBED=768, HIDDEN=1536, STATE=16, RANK=48, K=4
// ---------------------------------------------------------------------------

typedef __attribute__((ext_vector_type(2))) float v2f;
typedef __attribute__((ext_vector_type(8))) float v8f;

#define BATCH   4
#define SEQL    512
#define EMBED   768
#define HIDDEN  1536
#define STATE   16
#define RANKD   48
#define KW      4
#define ROWS    (BATCH * SEQL)      // 2048
#define BCDW    (RANKD + 2 * STATE) // 80

__device__ __forceinline__ float silu_f(float v) {
    return v / (1.0f + expf(-v));
}

__device__ __forceinline__ v8f wmma_f32_4(v2f a, v2f b, v8f c) {
    // D = A(16x4,f32) x B(4x16,f32) + C(16x16,f32) — CDNA5 V_WMMA_F32_16X16X4_F32
    return __builtin_amdgcn_wmma_f32_16x16x4_f32(
        /*neg_a=*/false, a, /*neg_b=*/false, b,
        /*c_mod=*/(short)0, c, /*reuse_a=*/false, /*reuse_b=*/false);
}

// ---------------------------------------------------------------------------
// Generic fp32 WMMA GEMM: C[M,N] = epi( A[M,K] @ B[K,N] )
// One wave computes an (MT*16) x (NT*16) macro-tile; 8 waves per block.
// A fragments loaded as 8B vectors; MT*NT independent accumulators keep the
// XDL pipe busy without D->A/B hazards.
// EPI 0: plain store.  EPI 1: softplus(acc + bias[col])  (delta path)
// ---------------------------------------------------------------------------
template <int EPI, int MT, int NT>
__global__ __launch_bounds__(256) void gemm_wmma_f32(
    const float* __restrict__ A, int lda,
    const float* __restrict__ B, int ldb,
    float* __restrict__ C, int ldc,
    int M, int N, int Kd,
    const float* __restrict__ bias)
{
    const int tid   = threadIdx.x;
    const int lane  = tid & 31;
    const int wave  = tid >> 5;
    const int ntn   = N / (16 * NT);                 // macro-tiles along N
    const int ntm   = M / (16 * MT);                 // macro-tiles along M
    const int tile  = blockIdx.x * 8 + wave;
    if (tile >= ntm * ntn) return;                   // uniform per wave
    const int m0 = (tile / ntn) * (16 * MT);
    const int n0 = (tile % ntn) * (16 * NT);
    const int r  = lane & 15;                        // A row in tile, B col in tile
    const int hi = lane >> 4;                        // K half selector

    v8f acc[MT][NT];
#pragma unroll
    for (int i = 0; i < MT; ++i)
#pragma unroll
        for (int j = 0; j < NT; ++j) acc[i][j] = {};

    const float* Arow[MT];
#pragma unroll
    for (int i = 0; i < MT; ++i)
        Arow[i] = A + (size_t)(m0 + i * 16 + r) * lda;

    for (int k0 = 0; k0 < Kd; k0 += 4) {
        const int ka = k0 + (hi << 1);
        v2f a[MT], b[NT];
#pragma unroll
        for (int i = 0; i < MT; ++i)
            a[i] = *reinterpret_cast<const v2f*>(Arow[i] + ka);   // global_load_b64
#pragma unroll
        for (int j = 0; j < NT; ++j) {
            b[j].x = B[(size_t)ka * ldb + n0 + j * 16 + r];
            b[j].y = B[(size_t)(ka + 1) * ldb + n0 + j * 16 + r];
        }
#pragma unroll
        for (int i = 0; i < MT; ++i)
#pragma unroll
            for (int j = 0; j < NT; ++j)
                acc[i][j] = wmma_f32_4(a[i], b[j], acc[i][j]);
    }

#pragma unroll
    for (int i = 0; i < MT; ++i) {
#pragma unroll
        for (int j = 0; j < NT; ++j) {
#pragma unroll
            for (int jj = 0; jj < 8; ++jj) {
                const int row = m0 + i * 16 + jj + (hi << 3);
                const int col = n0 + j * 16 + r;
                float v = acc[i][j][jj];
                if (EPI == 1) {
                    v += bias[col];
                    v = (v > 20.0f) ? v : log1pf(expf(v));   // softplus
                }
                C[(size_t)row * ldc + col] = v;
            }
        }
    }
}

// ---------------------------------------------------------------------------
// Conv-as-GEMM (the reference's faithful "time=hidden, chan=seq" Conv1D):
//   s[b,l,d] = silu( sum_{k,t} stream[b,t,d+k-3] * W_conv[k,t,l] + b_conv[l] )
// stream = xp[:, :, 0:HIDDEN].  Output written transposed into (b,l,d) layout.
// M = HIDDEN (d), N = SEQL (l), K = KW*SEQL (k,t).  32x64 macro-tile per wave.
// Causal zero-pad handled branch-free via clamped index + mask multiply.
// ---------------------------------------------------------------------------
__global__ __launch_bounds__(256) void conv_gemm_wmma(
    const float* __restrict__ xp,     // (ROWS, 2*HIDDEN)
    const float* __restrict__ Wconv,  // (KW, SEQL, HIDDEN)
    const float* __restrict__ bconv,  // (HIDDEN,) — first SEQL entries used
    float* __restrict__ S)            // (b,l,d) post-SiLU
{
    const int b    = blockIdx.y;
    const int tid  = threadIdx.x;
    const int lane = tid & 31;
    const int wave = tid >> 5;
    const int ntn  = SEQL / 64;                 // 8 macro-tiles along N(l)
    const int tile = blockIdx.x * 8 + wave;     // 48*8 = 384 macro-tiles
    const int m0 = (tile / ntn) * 32;           // d macro-tile
    const int n0 = (tile % ntn) * 64;           // l macro-tile
    const int r  = lane & 15;
    const int hi = lane >> 4;

    const float* Xb = xp + (size_t)b * SEQL * (2 * HIDDEN);
    v8f acc[2][4];
#pragma unroll
    for (int i = 0; i < 2; ++i)
#pragma unroll
        for (int j = 0; j < 4; ++j) acc[i][j] = {};

    for (int k = 0; k < KW; ++k) {
        const float* Bk = Wconv + (size_t)k * SEQL * HIDDEN;
        int   idx[2];
        float msk[2];
#pragma unroll
        for (int i = 0; i < 2; ++i) {
            const int dd = m0 + i * 16 + r + k - (KW - 1);  // shifted hidden index
            msk[i] = (dd >= 0) ? 1.0f : 0.0f;               // causal zero-pad
            idx[i] = (dd >= 0) ? dd : 0;
        }
        for (int t0 = 0; t0 < SEQL; t0 += 4) {
            const int ta = t0 + (hi << 1);
            v2f a[2], bb[4];
#pragma unroll
            for (int i = 0; i < 2; ++i) {
                a[i].x = Xb[(size_t)ta * (2 * HIDDEN) + idx[i]] * msk[i];
                a[i].y = Xb[(size_t)(ta + 1) * (2 * HIDDEN) + idx[i]] * msk[i];
            }
#pragma unroll
            for (int j = 0; j < 4; ++j) {
                bb[j].x = Bk[(size_t)ta * HIDDEN + n0 + j * 16 + r];
                bb[j].y = Bk[(size_t)(ta + 1) * HIDDEN + n0 + j * 16 + r];
            }
#pragma unroll
            for (int i = 0; i < 2; ++i)
#pragma unroll
                for (int j = 0; j < 4; ++j)
                    acc[i][j] = wmma_f32_4(a[i], bb[j], acc[i][j]);
        }
    }

#pragma unroll
    for (int i = 0; i < 2; ++i) {
#pragma unroll
        for (int j = 0; j < 4; ++j) {
#pragma unroll
            for (int jj = 0; jj < 8; ++jj) {
                const int drow = m0 + i * 16 + jj + (hi << 3);
                const int l    = n0 + j * 16 + r;
                const float v  = acc[i][j][jj] + bconv[l];
                S[((size_t)b * SEQL + l) * HIDDEN + drow] = silu_f(v);
            }
        }
    }
}

// ---------------------------------------------------------------------------
// Selective scan + fused output gating:
//   h_l = exp(delta*A) * h_{l-1} + delta*B_l*x ;  y_l = <h_l, C_l>
//   G[b,l,d] = (y + x*D[d]) * silu(res[b,l,d])
// One thread per (b,d); B_l/C_l (32 floats, shared by all d) staged via LDS.
// ---------------------------------------------------------------------------
__global__ __launch_bounds__(256) void scan_kernel(
    const float* __restrict__ S,      // (b,l,HIDDEN)
    const float* __restrict__ delta,  // (b,l,HIDDEN)
    const float* __restrict__ BCD,    // (b,l,BCDW): [0:48)=dr, [48:64)=B, [64:80)=C
    const float* __restrict__ xp,     // res at column offset HIDDEN
    const float* __restrict__ logA,   // (HIDDEN, STATE)
    const float* __restrict__ Dvec,   // (HIDDEN,)
    float* __restrict__ G)            // (b,l,HIDDEN)
{
    const int b = blockIdx.y;
    const int d = blockIdx.x * 256 + threadIdx.x;   // < HIDDEN

    float Arow[STATE], h[STATE];
#pragma unroll
    for (int n = 0; n < STATE; ++n) {
        Arow[n] = -expf(logA[(size_t)d * STATE + n]);
        h[n]    = 0.0f;
    }
    const float Dd = Dvec[d];

    __shared__ float sBC[2 * STATE];
    for (int l = 0; l < SEQL; ++l) {
        const size_t row = (size_t)b * SEQL + l;
        if (threadIdx.x < 2 * STATE)
            sBC[threadIdx.x] = BCD[row * BCDW + RANKD + threadIdx.x];
        __syncthreads();

        const float dl = delta[row * HIDDEN + d];
        const float x  = S[row * HIDDEN + d];
        const float dx = dl * x;
        float y = 0.0f;
#pragma unroll
        for (int n = 0; n < STATE; ++n) {
            h[n] = expf(dl * Arow[n]) * h[n] + dx * sBC[n];
            y   += h[n] * sBC[STATE + n];
        }
        const float res = xp[row * (2 * HIDDEN) + HIDDEN + d];
        G[row * HIDDEN + d] = (y + x * Dd) * silu_f(res);
        __syncthreads();   // protect sBC for next step
    }
}

// ---------------------------------------------------------------------------
extern "C" void kernel_launch(void* const* d_in, const int* in_sizes, int n_in,
                              void* d_out, int out_size, void* d_ws, size_t ws_size,
                              hipStream_t stream)
{
    (void)in_sizes; (void)n_in; (void)out_size; (void)ws_size;

    const float* inputs  = (const float*)d_in[0];
    const float* W_in    = (const float*)d_in[1];
    const float* W_conv  = (const float*)d_in[2];
    const float* b_conv  = (const float*)d_in[3];
    const float* W_param = (const float*)d_in[4];
    const float* W_delta = (const float*)d_in[5];
    const float* b_delta = (const float*)d_in[6];
    const float* log_A   = (const float*)d_in[7];
    const float* Dv      = (const float*)d_in[8];
    const float* W_out   = (const float*)d_in[9];
    float* out = (float*)d_out;

    // workspace layout (floats)
    float* xp  = (float*)d_ws;                       // ROWS * 2*HIDDEN  (stream|res)
    float* S   = xp  + (size_t)ROWS * 2 * HIDDEN;    // ROWS * HIDDEN
    float* dlt = S   + (size_t)ROWS * HIDDEN;        // ROWS * HIDDEN
    float* bcd = dlt + (size_t)ROWS * HIDDEN;        // ROWS * BCDW
    float* G   = bcd + (size_t)ROWS * BCDW;          // ROWS * HIDDEN

    // 1) xp = inputs @ W_in          (2048x768)@(768x3072), 32x64 macro-tiles
    gemm_wmma_f32<0, 2, 4><<<dim3((ROWS / 32) * ((2 * HIDDEN) / 64) / 8), 256, 0, stream>>>(
        inputs, EMBED, W_in, 2 * HIDDEN, xp, 2 * HIDDEN, ROWS, 2 * HIDDEN, EMBED, nullptr);

    // 2) conv (4 shifted GEMMs) + bias + SiLU -> S (b,l,d)
    conv_gemm_wmma<<<dim3((HIDDEN / 32) * (SEQL / 64) / 8, BATCH), 256, 0, stream>>>(
        xp, W_conv, b_conv, S);

    // 3) BCD = S @ W_param           (2048x1536)@(1536x80), 32x16 macro-tiles
    gemm_wmma_f32<0, 2, 1><<<dim3((ROWS / 32) * (BCDW / 16) / 8), 256, 0, stream>>>(
        S, HIDDEN, W_param, BCDW, bcd, BCDW, ROWS, BCDW, HIDDEN, nullptr);

    // 4) delta = softplus(BCD[:, :48] @ W_delta + b_delta)   (K=48)
    gemm_wmma_f32<1, 2, 4><<<dim3((ROWS / 32) * (HIDDEN / 64) / 8), 256, 0, stream>>>(
        bcd, BCDW, W_delta, HIDDEN, dlt, HIDDEN, ROWS, HIDDEN, RANKD, b_delta);

    // 5) selective scan + (y + x*D) * silu(res) -> G
    scan_kernel<<<dim3(HIDDEN / 256, BATCH), 256, 0, stream>>>(
        S, dlt, bcd, xp, log_A, Dv, G);

    // 6) out = G @ W_out             (2048x1536)@(1536x768)
    gemm_wmma_f32<0, 2, 4><<<dim3((ROWS / 32) * (EMBED / 64) / 8), 256, 0, stream>>>(
        G, HIDDEN, W_out, EMBED, out, EMBED, ROWS, EMBED, HIDDEN, nullptr);
}